// ContrastivePhaseObjective_28939489640867
// MI455X (gfx1250) — compile-verified
//
#include <hip/hip_runtime.h>
#include <math.h>

// ---------------- problem constants (match reference) ----------------
#define D        256          // embed dim
#define K2       512          // concat(real, imag) contraction length
#define MAXA     256          // MAX_ANCHORS
#define VOCABN   16000
#define TEMP     0.1f
#define MARGINF  1.0f
#define SEPW     1.0f
#define EPSF     1e-8f
#define BIGF     1000000000.0f

// ---------------- GEMM tiling ----------------
#define KC         32         // K-chunk staged in LDS
#define KCP        36         // padded LDS row stride (floats): conflict-free banks
#define NCHUNK     16         // K2 / KC
#define TSTEPS     8          // KC / 4 (k-step of the f32 16x16x4 WMMA)
#define MTILES     8          // anchor (M) tiles per block -> 128 anchors
#define AROWS      128        // anchors per block
#define NTILES     2          // column (N) tiles per wave -> 32 columns/wave
#define NCOL_BLOCK 256        // columns per block (8 waves * 32 cols)

typedef float v2f __attribute__((ext_vector_type(2)));
typedef float v8f __attribute__((ext_vector_type(8)));

// Monotone order-preserving float<->uint mapping so integer atomicMax/Min
// implement float max/min deterministically.
__device__ __forceinline__ unsigned encf(float f) {
  unsigned u = __float_as_uint(f);
  return (u & 0x80000000u) ? ~u : (u | 0x80000000u);
}
__device__ __forceinline__ float decf(unsigned u) {
  unsigned v = (u & 0x80000000u) ? (u ^ 0x80000000u) : ~u;
  return __uint_as_float(v);
}

// ---------------- kernel 1: init workspace accumulators ----------------
__global__ void k_init(int* counts, unsigned* posU, unsigned* negU, int* cnt) {
  int i = blockIdx.x * blockDim.x + threadIdx.x;
  if (i < VOCABN) counts[i] = 0;
  if (i < MAXA) {
    posU[i] = encf(-BIGF);   // matches reference where(valid, sims, -BIG)
    negU[i] = encf( BIGF);
    cnt[i]  = 0;
  }
}

// ---------------- kernel 2: token histogram ----------------
__global__ void k_hist(const int* __restrict__ tok, int* counts, int N) {
  int i = blockIdx.x * blockDim.x + threadIdx.x;
  if (i < N) atomicAdd(&counts[tok[i]], 1);
}

// ---------------- kernel 3: stable anchor selection ----------------
// Reference: stable argsort on (~repeated) -> first MAXA indices.
// Equivalent: first the repeated indices in index order, then non-repeated.
__global__ void k_select(const int* __restrict__ tok, const int* __restrict__ counts,
                         int* aidx, int* aok, int N) {
  __shared__ int s_base;
  __shared__ int s_wcnt[8];
  int tid = threadIdx.x;
  int lane = tid & 31, wave = tid >> 5;
  unsigned lmask = (1u << lane) - 1u;

  if (tid == 0) s_base = 0;
  __syncthreads();

  // pass 1: repeated tokens, in index order
  for (int chunk = 0; chunk < N; chunk += 256) {
    int i = chunk + tid;
    bool rep = counts[tok[i]] >= 2;
    unsigned mask = (unsigned)__ballot(rep);
    if (lane == 0) s_wcnt[wave] = __popc(mask);
    __syncthreads();
    int prefix = 0;
    for (int w = 0; w < wave; ++w) prefix += s_wcnt[w];
    int pos = s_base + prefix + __popc(mask & lmask);
    if (rep && pos < MAXA) { aidx[pos] = i; aok[pos] = 1; }
    __syncthreads();
    if (tid == 0) { int t = 0; for (int w = 0; w < 8; ++w) t += s_wcnt[w]; s_base += t; }
    __syncthreads();
    if (s_base >= MAXA) break;
  }
  __syncthreads();
  int filled = min(s_base, MAXA);
  __syncthreads();
  if (filled >= MAXA) return;

  if (tid == 0) s_base = 0;
  __syncthreads();

  // pass 2: fill remaining slots with non-repeated indices, in index order
  for (int chunk = 0; chunk < N; chunk += 256) {
    int i = chunk + tid;
    bool sel = counts[tok[i]] < 2;
    unsigned mask = (unsigned)__ballot(sel);
    if (lane == 0) s_wcnt[wave] = __popc(mask);
    __syncthreads();
    int prefix = 0;
    for (int w = 0; w < wave; ++w) prefix += s_wcnt[w];
    int pos = filled + s_base + prefix + __popc(mask & lmask);
    if (sel && pos < MAXA) { aidx[pos] = i; aok[pos] = 0; }
    __syncthreads();
    if (tid == 0) { int t = 0; for (int w = 0; w < 8; ++w) t += s_wcnt[w]; s_base += t; }
    __syncthreads();
    if (filled + s_base >= MAXA) break;
  }
}

// ---------------- kernel 4: gather anchor rows, anchor mags/tokens ----------------
__global__ void k_pack(const float* __restrict__ R, const float* __restrict__ I,
                       const int* __restrict__ tok, const int* __restrict__ aidx,
                       float* Apack, float* ma, int* ta) {
  int a = blockIdx.x;
  int t = threadIdx.x;
  int row = aidx[a];
  float v0 = R[(size_t)row * D + t];
  float v1 = I[(size_t)row * D + t];
  Apack[(size_t)a * K2 + t]     = v0;
  Apack[(size_t)a * K2 + D + t] = v1;
  __shared__ float red[256];
  red[t] = v0 * v0 + v1 * v1;
  __syncthreads();
  for (int s = 128; s > 0; s >>= 1) {
    if (t < s) red[t] += red[t + s];
    __syncthreads();
  }
  if (t == 0) { ma[a] = sqrtf(red[0] + EPSF); ta[a] = tok[row]; }
}

// ---------------- kernel 5: fused WMMA GEMM + sims + per-anchor max/min/count ----
// Block = 256 threads (8 waves). Wave w owns 2 N-tiles (32 columns); block owns
// 256 columns x 128 anchors (gridDim.y=2 selects the anchor half). A chunks are
// staged in LDS with a padded row stride (KCP=36 floats) so the per-lane
// fragment reads hit 32 distinct banks (the unpadded stride-32 layout was
// ~8-way conflicted). Each A fragment feeds TWO independent WMMAs (both
// N-tiles), and the next fragment group's ds_load is in flight while the
// current WMMA group occupies the matrix pipe. Column magnitudes are computed
// from the streamed B data (no separate mag pass over HBM).
__global__ void __launch_bounds__(256, 1)
k_gemm(const float* __restrict__ R, const float* __restrict__ I,
       const int* __restrict__ tok, const float* __restrict__ Apack,
       const float* __restrict__ ma, const int* __restrict__ ta,
       const int* __restrict__ aidx,
       unsigned* posU, unsigned* negU, int* cnt) {
  __shared__ float    sA[AROWS * KCP];  // 18 KB padded A chunk
  __shared__ unsigned sPos[AROWS];
  __shared__ unsigned sNeg[AROWS];
  __shared__ int      sCnt[AROWS];
  __shared__ int      sTa[AROWS];
  __shared__ int      sAidx[AROWS];
  __shared__ float    sMa[AROWS];

  int tid  = threadIdx.x;
  int lane = tid & 31, wave = tid >> 5;
  int h  = lane >> 4;        // lane half (wave32: 2 halves of 16)
  int nn = lane & 15;        // lane-within-half: column / anchor id inside tile
  int abase   = blockIdx.y * AROWS;
  int colbase = blockIdx.x * NCOL_BLOCK;
  int col0 = colbase + wave * 32 + nn;   // N-tile 0
  int col1 = col0 + 16;                  // N-tile 1

  if (tid < AROWS) {
    sPos[tid] = encf(-BIGF);
    sNeg[tid] = encf( BIGF);
    sCnt[tid] = 0;
    sTa[tid]   = ta[abase + tid];
    sAidx[tid] = aidx[abase + tid];
    sMa[tid]   = ma[abase + tid];
  }

  v8f acc[MTILES][NTILES];
  v8f zero = {0.f, 0.f, 0.f, 0.f, 0.f, 0.f, 0.f, 0.f};
#pragma unroll
  for (int m = 0; m < MTILES; ++m) {
    acc[m][0] = zero;
    acc[m][1] = zero;
  }
  float ssq0 = 0.0f, ssq1 = 0.0f;

  __syncthreads();

  for (int kc = 0; kc < NCHUNK; ++kc) {
    // cooperatively stage A chunk: Apack[abase:abase+128][kc*KC .. +KC] -> LDS
#pragma unroll
    for (int u = 0; u < 4; ++u) {
      int idx  = tid + u * 256;        // 0..1023 float4 slots
      int arow = idx >> 3;             // 0..127
      int k4   = idx & 7;
      float4 g = *(const float4*)(Apack + (size_t)(abase + arow) * K2 + kc * KC + k4 * 4);
      *(float4*)(sA + arow * KCP + k4 * 4) = g;
    }
    // load B fragments for this wave's two columns (contiguous along K)
    const float* base = (kc < 8 ? R : I);
    const float* cp0 = base + (size_t)col0 * D + (kc & 7) * KC;
    const float* cp1 = base + (size_t)col1 * D + (kc & 7) * KC;
    float2 bf0[TSTEPS], bf1[TSTEPS];
#pragma unroll
    for (int ts = 0; ts < TSTEPS; ++ts) {
      bf0[ts] = *(const float2*)(cp0 + 4 * ts + 2 * h);
      bf1[ts] = *(const float2*)(cp1 + 4 * ts + 2 * h);
      ssq0 += bf0[ts].x * bf0[ts].x + bf0[ts].y * bf0[ts].y;
      ssq1 += bf1[ts].x * bf1[ts].x + bf1[ts].y * bf1[ts].y;
    }
    __syncthreads();

    // A fragments across M-tiles; each frag feeds 2 WMMAs (both N-tiles)
#pragma unroll
    for (int m = 0; m < MTILES; ++m) {
      const float* ap = sA + (m * 16 + nn) * KCP + 2 * h;
      v2f af[TSTEPS];
#pragma unroll
      for (int ts = 0; ts < TSTEPS; ++ts) {
        af[ts].x = ap[4 * ts];
        af[ts].y = ap[4 * ts + 1];
      }
#pragma unroll
      for (int ts = 0; ts < TSTEPS; ++ts) {
        v2f b0; b0.x = bf0[ts].x; b0.y = bf0[ts].y;
        v2f b1; b1.x = bf1[ts].x; b1.y = bf1[ts].y;
        acc[m][0] = __builtin_amdgcn_wmma_f32_16x16x4_f32(
            false, af[ts], false, b0, (short)0, acc[m][0], false, false);
        acc[m][1] = __builtin_amdgcn_wmma_f32_16x16x4_f32(
            false, af[ts], false, b1, (short)0, acc[m][1], false, false);
      }
    }
    __syncthreads();
  }

  // full ||x_col||: each lane-half summed half the K values
  float mag0 = sqrtf(ssq0 + __shfl_xor(ssq0, 16) + EPSF);
  float mag1 = sqrtf(ssq1 + __shfl_xor(ssq1, 16) + EPSF);
  int tok0 = tok[col0];
  int tok1 = tok[col1];

  // epilogue: sims + validity + half-wave reductions + LDS atomics
#pragma unroll
  for (int m = 0; m < MTILES; ++m) {
#pragma unroll
    for (int r = 0; r < 8; ++r) {
      int row = m * 16 + r + 8 * h;           // local anchor index (0..127)
      float rm = sMa[row];
      int   rt = sTa[row];
      int   ri = sAidx[row];
#pragma unroll
      for (int j = 0; j < NTILES; ++j) {
        float magc = (j == 0) ? mag0 : mag1;
        int   tokc = (j == 0) ? tok0 : tok1;
        int   colj = (j == 0) ? col0 : col1;
        float sim = acc[m][j][r] / (rm * magc + EPSF);
        bool valid = (rt == tokc) && (ri != colj);
        float pv = valid ? sim : -BIGF;
        float nv = valid ? sim :  BIGF;
        int   cv = valid ? 1 : 0;
#pragma unroll
        for (int o = 1; o < 16; o <<= 1) {    // reduce 16 columns sharing a row
          pv = fmaxf(pv, __shfl_xor(pv, o));
          nv = fminf(nv, __shfl_xor(nv, o));
          cv += __shfl_xor(cv, o);
        }
        if (nn == 0 && cv > 0) {
          atomicMax(&sPos[row], encf(pv));
          atomicMin(&sNeg[row], encf(nv));
          atomicAdd(&sCnt[row], cv);
        }
      }
    }
  }
  __syncthreads();

  // one global atomic triple per anchor per block
  if (tid < AROWS && sCnt[tid] > 0) {
    atomicMax(&posU[abase + tid], sPos[tid]);
    atomicMin(&negU[abase + tid], sNeg[tid]);
    atomicAdd(&cnt[abase + tid],  sCnt[tid]);
  }
}

// ---------------- kernel 6: loss ----------------
__global__ void k_final(const unsigned* __restrict__ posU, const unsigned* __restrict__ negU,
                        const int* __restrict__ cnt, const int* __restrict__ aok,
                        float* out) {
  __shared__ float sTot[256];
  __shared__ int   sNum[256];
  int t = threadIdx.x;
  float pos = decf(posU[t]);
  float neg = decf(negU[t]);
  int num = cnt[t];
  int ok = (aok[t] != 0) && (num >= 2);
  float l0 = pos / TEMP, l1 = neg / TEMP;
  float mx = fmaxf(l0, l1);
  float lse = mx + logf(expf(l0 - mx) + expf(l1 - mx));
  float ce  = lse - l0;
  float sep = fmaxf(neg + MARGINF, 0.0f);
  float pa  = ce + SEPW * sep;
  sTot[t] = ok ? pa : 0.0f;
  sNum[t] = ok ? 1 : 0;
  __syncthreads();
  for (int s = 128; s > 0; s >>= 1) {
    if (t < s) { sTot[t] += sTot[t + s]; sNum[t] += sNum[t + s]; }
    __syncthreads();
  }
  if (t == 0) out[0] = (sNum[0] > 0) ? sTot[0] / (float)sNum[0] : 0.0f;
}

// ---------------- launcher ----------------
extern "C" void kernel_launch(void* const* d_in, const int* in_sizes, int n_in,
                              void* d_out, int out_size, void* d_ws, size_t ws_size,
                              hipStream_t stream) {
  const float* R  = (const float*)d_in[0];
  const float* I  = (const float*)d_in[1];
  const int*  tok = (const int*)d_in[2];
  int N = in_sizes[2];                    // B*S = 65536

  char* ws = (char*)d_ws;
  int*      counts = (int*)      (ws + 0);              // 16000 ints (64000 B)
  int*      aidx   = (int*)      (ws + 65536);          // 256
  int*      aok    = (int*)      (ws + 65536 + 1024);   // 256
  int*      ta     = (int*)      (ws + 65536 + 2048);   // 256
  float*    ma     = (float*)    (ws + 65536 + 3072);   // 256
  unsigned* posU   = (unsigned*) (ws + 65536 + 4096);   // 256
  unsigned* negU   = (unsigned*) (ws + 65536 + 5120);   // 256
  int*      cnt    = (int*)      (ws + 65536 + 6144);   // 256
  float*    Apack  = (float*)    (ws + 65536 + 8192);   // 256*512 floats = 512 KB

  k_init  <<<(VOCABN + 255) / 256, 256, 0, stream>>>(counts, posU, negU, cnt);
  k_hist  <<<(N + 255) / 256,      256, 0, stream>>>(tok, counts, N);
  k_select<<<1,                    256, 0, stream>>>(tok, counts, aidx, aok, N);
  k_pack  <<<MAXA,                 256, 0, stream>>>(R, I, tok, aidx, Apack, ma, ta);
  dim3 ggrid(N / NCOL_BLOCK, MAXA / AROWS);
  k_gemm  <<<ggrid,                256, 0, stream>>>(R, I, tok, Apack, ma, ta, aidx,
                                                     posU, negU, cnt);
  k_final <<<1,                    256, 0, stream>>>(posU, negU, cnt, aok, (float*)d_out);
}